// Sage_77481210020254
// MI455X (gfx1250) — compile-verified
//
#include <hip/hip_runtime.h>

typedef __attribute__((ext_vector_type(2))) float v2f;
typedef __attribute__((ext_vector_type(8))) float v8f;

#define HID 64
#define EPS 1e-5f

// =====================================================================
// WMMA GEMM: out[M,64] = act(A[M,64] @ W[64,64] + bias[64])
// One wave computes a 16x64 strip via V_WMMA_F32_16X16X4_F32.
// A-frag (16x4 f32): lane l -> m = l&15, k = k0 + 2*(l>>4) + {0,1}
// B-frag (4x16 f32): lane l -> n = l&15, k = k0 + 2*(l>>4) + vgpr
// C/D  (16x16 f32): lane l, vgpr r -> m = r + 8*(l>>4), n = l&15
// =====================================================================
template <bool RELU>
__global__ __launch_bounds__(256) void gemm64_kernel(const float* __restrict__ A,
                                                     const float* __restrict__ W,
                                                     const float* __restrict__ bias,
                                                     float* __restrict__ out, int M) {
    const int wave = threadIdx.x >> 5;
    const int lane = threadIdx.x & 31;
    const int tile = blockIdx.x * 8 + wave;
    if (tile * 16 >= M) return;              // wave-uniform: EXEC stays all-ones
    const int up  = lane >> 4;
    const int l15 = lane & 15;

    const float* __restrict__ arow = A + (size_t)(tile * 16 + l15) * HID;
    v8f acc[4] = {};

    for (int k0 = 0; k0 < HID; k0 += 4) {
        const int ka = k0 + (up << 1);
        v2f a = *(const v2f*)(arow + ka);
        const float* __restrict__ wk = W + (size_t)ka * HID;
#pragma unroll
        for (int nt = 0; nt < 4; ++nt) {
            v2f b;
            b.x = wk[nt * 16 + l15];
            b.y = wk[HID + nt * 16 + l15];
            acc[nt] = __builtin_amdgcn_wmma_f32_16x16x4_f32(
                false, a, false, b, (short)0, acc[nt], false, false);
        }
    }

#pragma unroll
    for (int nt = 0; nt < 4; ++nt) {
        const int col = nt * 16 + l15;
        const float bv = bias[col];
#pragma unroll
        for (int r = 0; r < 8; ++r) {
            float v = acc[nt][r] + bv;
            if (RELU) v = fmaxf(v, 0.0f);
            out[(size_t)(tile * 16 + r + (up << 3)) * HID + col] = v;
        }
    }
}

// =====================================================================
// SAGE dual GEMM: h[M,64] = relu((agg*inv) @ Wl + bl + h @ Wr)
// Two WMMA streams chained into one accumulator. In-place on h is safe:
// each wave reads only the 16 rows it later writes.
// =====================================================================
__global__ __launch_bounds__(256) void sage_gemm_kernel(const float* __restrict__ agg,
                                                        float* __restrict__ h,
                                                        const float* __restrict__ inv,
                                                        const float* __restrict__ Wl,
                                                        const float* __restrict__ Wr,
                                                        const float* __restrict__ bl,
                                                        int M) {
    const int wave = threadIdx.x >> 5;
    const int lane = threadIdx.x & 31;
    const int tile = blockIdx.x * 8 + wave;
    if (tile * 16 >= M) return;
    const int up  = lane >> 4;
    const int l15 = lane & 15;

    const float invm = inv[tile * 16 + l15];
    const float* __restrict__ grow = agg + (size_t)(tile * 16 + l15) * HID;
    const float* __restrict__ hrow = h   + (size_t)(tile * 16 + l15) * HID;
    v8f acc[4] = {};

    for (int k0 = 0; k0 < HID; k0 += 4) {
        const int ka = k0 + (up << 1);
        v2f ag = *(const v2f*)(grow + ka);
        ag.x *= invm; ag.y *= invm;
        v2f ah = *(const v2f*)(hrow + ka);
        const float* __restrict__ wlk = Wl + (size_t)ka * HID;
        const float* __restrict__ wrk = Wr + (size_t)ka * HID;
#pragma unroll
        for (int nt = 0; nt < 4; ++nt) {
            v2f bl2, br2;
            bl2.x = wlk[nt * 16 + l15];
            bl2.y = wlk[HID + nt * 16 + l15];
            br2.x = wrk[nt * 16 + l15];
            br2.y = wrk[HID + nt * 16 + l15];
            acc[nt] = __builtin_amdgcn_wmma_f32_16x16x4_f32(
                false, ag, false, bl2, (short)0, acc[nt], false, false);
            acc[nt] = __builtin_amdgcn_wmma_f32_16x16x4_f32(
                false, ah, false, br2, (short)0, acc[nt], false, false);
        }
    }

#pragma unroll
    for (int nt = 0; nt < 4; ++nt) {
        const int col = nt * 16 + l15;
        const float bv = bl[col];
#pragma unroll
        for (int r = 0; r < 8; ++r) {
            float v = fmaxf(acc[nt][r] + bv, 0.0f);
            h[(size_t)(tile * 16 + r + (up << 3)) * HID + col] = v;
        }
    }
}

// ======================= LayerNorm over 64 features ===================
__device__ __forceinline__ float wave_sum(float v) {
#pragma unroll
    for (int o = 16; o > 0; o >>= 1) v += __shfl_xor(v, o, 32);
    return v;
}

__global__ __launch_bounds__(256) void layernorm_kernel(const float* __restrict__ in,
                                                        float* __restrict__ out,
                                                        const float* __restrict__ g,
                                                        const float* __restrict__ b, int M) {
    const int wave = threadIdx.x >> 5;
    const int lane = threadIdx.x & 31;
    const int row  = blockIdx.x * 8 + wave;
    if (row >= M) return;
    const float* p = in + (size_t)row * HID;
    float x0 = p[lane], x1 = p[lane + 32];
    float mean = wave_sum(x0 + x1) * (1.0f / HID);
    float d0 = x0 - mean, d1 = x1 - mean;
    float var = wave_sum(d0 * d0 + d1 * d1) * (1.0f / HID);
    float rstd = rsqrtf(var + EPS);
    float* q = out + (size_t)row * HID;
    q[lane]      = d0 * rstd * g[lane]      + b[lane];
    q[lane + 32] = d1 * rstd * g[lane + 32] + b[lane + 32];
}

// ======================= degree count / inverse =======================
__global__ void count_kernel(const int* __restrict__ dst, int* __restrict__ cnt, int E) {
    int e = blockIdx.x * 256 + threadIdx.x;
    if (e < E) atomicAdd(&cnt[dst[e]], 1);
}

__global__ void inv_kernel(float* __restrict__ buf, int M) {
    int i = blockIdx.x * 256 + threadIdx.x;
    if (i < M) {
        int c = ((const int*)buf)[i];
        buf[i] = 1.0f / (float)(c > 1 ? c : 1);
    }
}

// =========== edge scatter-add: agg[dst] += h[src]  (L2-resident) ======
__global__ __launch_bounds__(256) void scatter_kernel(const int* __restrict__ src,
                                                      const int* __restrict__ dst,
                                                      const float* __restrict__ h,
                                                      float* __restrict__ agg, int E) {
    long long gid = (long long)blockIdx.x * 256 + threadIdx.x;
    int e = (int)(gid >> 4);
    if (e >= E) return;
    int part = ((int)gid & 15) << 2;
    const float4 v = *(const float4*)(h + (size_t)src[e] * HID + part);
    float* ap = agg + (size_t)dst[e] * HID + part;
    atomicAdd(ap + 0, v.x);
    atomicAdd(ap + 1, v.y);
    atomicAdd(ap + 2, v.z);
    atomicAdd(ap + 3, v.w);
}

// ======================= global add-pool ==============================
__global__ void pool_kernel(const float* __restrict__ h, const int* __restrict__ batch,
                            float* __restrict__ g, int M) {
    long long gid = (long long)blockIdx.x * 256 + threadIdx.x;
    if (gid >= (long long)M * HID) return;
    int node = (int)(gid >> 6), c = (int)gid & 63;
    atomicAdd(&g[(size_t)batch[node] * HID + c], h[gid]);
}

// ======================= per-graph decoders ===========================
struct DecParams {
    const float *W0, *b0, *W1, *b1, *lng0, *lnb0, *lng1, *lnb1;
};

__global__ __launch_bounds__(256) void decoder_kernel(const float* __restrict__ g,
                                                      DecParams p0, DecParams p1,
                                                      float* __restrict__ out, int G) {
    __shared__ float buf[8][HID];
    const int wave = threadIdx.x >> 5;
    const int lane = threadIdx.x & 31;
    const int row  = blockIdx.x * 8 + wave;
    if (row >= G) return;
    const float x0 = g[(size_t)row * HID + lane];
    const float x1 = g[(size_t)row * HID + lane + 32];

#pragma unroll
    for (int d = 0; d < 2; ++d) {
        const DecParams P = d ? p1 : p0;
        // LN1
        float mean = wave_sum(x0 + x1) * (1.0f / HID);
        float d0 = x0 - mean, d1 = x1 - mean;
        float rstd = rsqrtf(wave_sum(d0 * d0 + d1 * d1) * (1.0f / HID) + EPS);
        buf[wave][lane]      = d0 * rstd * P.lng0[lane]      + P.lnb0[lane];
        buf[wave][lane + 32] = d1 * rstd * P.lng0[lane + 32] + P.lnb0[lane + 32];
        __builtin_amdgcn_wave_barrier();   // wave-coherent LDS, DS ops are in-order
        // Linear(64,64) + ReLU : each lane owns cols lane, lane+32
        float y0 = P.b0[lane], y1 = P.b0[lane + 32];
        for (int k = 0; k < HID; ++k) {
            float xv = buf[wave][k];
            y0 += xv * P.W0[k * HID + lane];
            y1 += xv * P.W0[k * HID + lane + 32];
        }
        y0 = fmaxf(y0, 0.0f);
        y1 = fmaxf(y1, 0.0f);
        // LN2
        float m2 = wave_sum(y0 + y1) * (1.0f / HID);
        float e0 = y0 - m2, e1 = y1 - m2;
        float rs2 = rsqrtf(wave_sum(e0 * e0 + e1 * e1) * (1.0f / HID) + EPS);
        float n0 = e0 * rs2 * P.lng1[lane]      + P.lnb1[lane];
        float n1 = e1 * rs2 * P.lng1[lane + 32] + P.lnb1[lane + 32];
        // Linear(64,1) + ReLU
        float s = wave_sum(n0 * P.W1[lane] + n1 * P.W1[lane + 32]);
        if (lane == 0) out[(size_t)row * 2 + d] = fmaxf(s + P.b1[0], 0.0f);
    }
}

// =====================================================================
extern "C" void kernel_launch(void* const* d_in, const int* in_sizes, int n_in,
                              void* d_out, int out_size, void* d_ws, size_t ws_size,
                              hipStream_t stream) {
    const int M = in_sizes[0] / HID;   // 100000 nodes
    const int E = in_sizes[1] / 2;     // 3.2M edges
    const int G = out_size / 2;        // 256 graphs

    const float* x    = (const float*)d_in[0];
    const int* eidx   = (const int*)d_in[1];   // [2,E]: src then dst
    const int* batch  = (const int*)d_in[2];
    // d_in[3] = num_graphs scalar (unused; G from out_size)

    // params flattened in jax tree order (dict keys sorted)
    int i = 4;
    const float *convWl[3], *convWr[3], *convbl[3];
    for (int c = 0; c < 3; ++c) {                    // convs: Wl < Wr < bl
        convWl[c] = (const float*)d_in[i++];
        convWr[c] = (const float*)d_in[i++];
        convbl[c] = (const float*)d_in[i++];
    }
    DecParams dp[2];
    for (int d = 0; d < 2; ++d) {                    // dec: W < b < ln_b < ln_g
        dp[d].W0   = (const float*)d_in[i++];
        dp[d].W1   = (const float*)d_in[i++];
        dp[d].b0   = (const float*)d_in[i++];
        dp[d].b1   = (const float*)d_in[i++];
        dp[d].lnb0 = (const float*)d_in[i++];
        dp[d].lnb1 = (const float*)d_in[i++];
        dp[d].lng0 = (const float*)d_in[i++];
        dp[d].lng1 = (const float*)d_in[i++];
    }
    const float *encW[4], *encb[4];                  // enc: Ws < bs < ln_b < ln_g
    for (int k = 0; k < 4; ++k) encW[k] = (const float*)d_in[i++];
    for (int k = 0; k < 4; ++k) encb[k] = (const float*)d_in[i++];
    const float* enc_lnb = (const float*)d_in[i++];
    const float* enc_lng = (const float*)d_in[i++];

    // workspace layout (256B-aligned chunks)
    const size_t hbytes = ((size_t)M * HID * 4 + 255) & ~(size_t)255;
    char* ws = (char*)d_ws;
    float* hA   = (float*)ws;                      // node features
    float* hB   = (float*)(ws + hbytes);           // ping-pong / agg
    float* gbuf = (float*)(ws + 2 * hbytes);       // [G,64] pooled
    float* dbuf = (float*)(ws + 2 * hbytes + (((size_t)G * HID * 4 + 255) & ~(size_t)255)); // degree

    const int MB  = (M + 127) / 128;               // GEMM blocks (8 waves x 16 rows)
    const int LNB = (M + 7) / 8;

    // ---- encoder MLP ----
    gemm64_kernel<true ><<<MB, 256, 0, stream>>>(x,  encW[0], encb[0], hA, M);
    gemm64_kernel<true ><<<MB, 256, 0, stream>>>(hA, encW[1], encb[1], hB, M);
    gemm64_kernel<true ><<<MB, 256, 0, stream>>>(hB, encW[2], encb[2], hA, M);
    layernorm_kernel<<<LNB, 256, 0, stream>>>(hA, hB, enc_lng, enc_lnb, M);
    gemm64_kernel<false><<<MB, 256, 0, stream>>>(hB, encW[3], encb[3], hA, M);

    // ---- degree -> 1/max(deg,1) ----
    hipMemsetAsync(dbuf, 0, (size_t)M * 4, stream);
    count_kernel<<<(E + 255) / 256, 256, 0, stream>>>(eidx + E, (int*)dbuf, E);
    inv_kernel<<<(M + 255) / 256, 256, 0, stream>>>(dbuf, M);

    // ---- SAGE convs ----
    const long long sth = (long long)E * 16;
    const int SB = (int)((sth + 255) / 256);
    for (int c = 0; c < 3; ++c) {
        hipMemsetAsync(hB, 0, (size_t)M * HID * 4, stream);
        scatter_kernel<<<SB, 256, 0, stream>>>(eidx, eidx + E, hA, hB, E);
        sage_gemm_kernel<<<MB, 256, 0, stream>>>(hB, hA, dbuf, convWl[c], convWr[c], convbl[c], M);
    }

    // ---- pool + decoders ----
    hipMemsetAsync(gbuf, 0, (size_t)G * HID * 4, stream);
    const long long pth = (long long)M * HID;
    pool_kernel<<<(int)((pth + 255) / 256), 256, 0, stream>>>(hA, batch, gbuf, M);
    decoder_kernel<<<(G + 7) / 8, 256, 0, stream>>>(gbuf, dp[0], dp[1], (float*)d_out, G);
}